// Net1_45896020525614
// MI455X (gfx1250) — compile-verified
//
#include <hip/hip_runtime.h>

typedef __attribute__((ext_vector_type(2))) float v2f;
typedef __attribute__((ext_vector_type(8))) float v8f;

#define TSTEPS 2048
#define BATCH  4096
#define FIN    2
#define HID    8
#define NOUT   4
#define NGATE  24   // 3*HID, PyTorch gate order [r, z, n]
#define LOG2E  1.44269504088896340736f

// ---------------------------------------------------------------------------
// V_WMMA_F32_16X16X4_F32 : D(16x16,f32) = A(16x4,f32) * B(4x16,f32) + C
// A: 2 VGPRs/lane (lanes 0-15: K=0,1 ; lanes 16-31: K=2,3), M = lane%16
// B: 2 VGPRs/lane (VGPR v: K = v + 2*(lane/16)),             N = lane%16
// C/D: 8 VGPRs/lane (VGPR v: M = v + 8*(lane/16)),           N = lane%16
// ---------------------------------------------------------------------------
static __device__ inline v8f wmma4(v2f a, v2f b, v8f c) {
  return __builtin_amdgcn_wmma_f32_16x16x4_f32(
      /*neg_a=*/false, a, /*neg_b=*/false, b,
      /*c_mod=*/(short)0, c, /*reuse_a=*/false, /*reuse_b=*/false);
}

static __device__ inline void wait_ds() {
  asm volatile("s_wait_dscnt 0" ::: "memory");
}

static __device__ inline float fast_exp2(float x) {
#if __has_builtin(__builtin_amdgcn_exp2f)
  return __builtin_amdgcn_exp2f(x);
#else
  return exp2f(x);
#endif
}
static __device__ inline float fast_rcp(float x) {
#if __has_builtin(__builtin_amdgcn_rcpf)
  return __builtin_amdgcn_rcpf(x);
#else
  return 1.f / x;
#endif
}
// sigmoid(x) from PRE-SCALED argument a = -log2(e)*x :
//   s = rcp(1 + exp2(a))          (exp2 + add + rcp; saturates correctly at +-inf)
static __device__ inline float sigm_s(float a) {
  return fast_rcp(1.f + fast_exp2(a));
}
// tanh(x) from PRE-SCALED argument a = -2*log2(e)*x :
//   t = 2*rcp(1 + exp2(a)) - 1    (exp2 + add + rcp + fma)
static __device__ inline float tanh_s(float a) {
  return __builtin_fmaf(2.f, fast_rcp(1.f + fast_exp2(a)), -1.f);
}

// lane j <-> lane j^8 within each 16-lane row: one v_permlane16_b32
static __device__ inline float lane_xor8(float v) {
#if __has_builtin(__builtin_amdgcn_permlane16)
  int i = __builtin_bit_cast(int, v);
  i = __builtin_amdgcn_permlane16(i, i, 0xFEDCBA98u, 0x76543210u, false, false);
  return __builtin_bit_cast(float, i);
#else
  return __shfl_xor(v, 8, 32);
#endif
}

static __device__ inline v8f splat8(float v) {
  v8f r = {v, v, v, v, v, v, v, v};
  return r;
}

// B-operand for W^T: tile selects gate columns [16*tile, 16*tile+16),
// khalf selects K-group (0: K=0..3, 1: K=4..7). Zero-padded outside (NGATE,Kdim).
static __device__ inline v2f loadB(const float* __restrict__ W, int Kdim,
                                   int lane, int tile, int khalf, float scale) {
  int g  = (lane & 15) + 16 * tile;
  int kb = ((lane >> 4) << 1) + (khalf << 2);
  v2f r = {0.f, 0.f};
  if (g < NGATE) {
    if (kb     < Kdim) r.x = scale * W[g * Kdim + kb];
    if (kb + 1 < Kdim) r.y = scale * W[g * Kdim + kb + 1];
  }
  return r;
}
static __device__ inline float loadBias(const float* __restrict__ b, int lane, int tile) {
  int g = (lane & 15) + 16 * tile;
  return (g < NGATE) ? b[g] : 0.f;
}

struct LayerW {
  v2f bi00, bi01, bi10, bi11;   // input-proj  B operands [tile][khalf] (pre-scaled)
  v2f bh00, bh01, bh10, bh11;   // hidden-proj B operands [tile][khalf] (pre-scaled)
  float cb1;                    // -log2e * (b_ih + b_hh) for tile0 ([r|z])
  float cbi1, cbh1;             // -2*log2e * biases for tile1 ([n|-])
};

// Activation argument scales folded into the weights: tile0 feeds only the
// sigmoid (scale -log2e), tile1 feeds only the tanh (scale -2*log2e).
static __device__ inline LayerW make_layer(const float* __restrict__ Wih, int Kin,
                                           const float* __restrict__ Whh,
                                           const float* __restrict__ bih,
                                           const float* __restrict__ bhh, int lane) {
  const float SG = -LOG2E;
  const float SN = -2.f * LOG2E;
  LayerW w;
  w.bi00 = loadB(Wih, Kin, lane, 0, 0, SG); w.bi01 = loadB(Wih, Kin, lane, 0, 1, SG);
  w.bi10 = loadB(Wih, Kin, lane, 1, 0, SN); w.bi11 = loadB(Wih, Kin, lane, 1, 1, SN);
  w.bh00 = loadB(Whh, HID, lane, 0, 0, SG); w.bh01 = loadB(Whh, HID, lane, 0, 1, SG);
  w.bh10 = loadB(Whh, HID, lane, 1, 0, SN); w.bh11 = loadB(Whh, HID, lane, 1, 1, SN);
  w.cb1  = SG * (loadBias(bih, lane, 0) + loadBias(bhh, lane, 0));
  w.cbi1 = SN * loadBias(bih, lane, 1);
  w.cbh1 = SN * loadBias(bhh, lane, 1);
  return w;
}

// One GRU step for a 16-row batch tile. Biases pre-seeded into the WMMA
// accumulators; HAS_HI=false elides the K=4..7 input-proj WMMAs (layer 0, F=2).
template <bool HAS_HI>
static __device__ inline v8f gru_step(const LayerW& w, v2f ai_lo, v2f ai_hi,
                                      v2f ah_lo, v2f ah_hi, v8f h) {
  // g1 = -log2e * (x[r|z] + h[r|z] + biases)   (single accumulator chain)
  v8f g1 = wmma4(ah_hi, w.bh01, splat8(w.cb1));
  g1 = wmma4(ah_lo, w.bh00, g1);
  g1 = wmma4(ai_lo, w.bi00, g1);
  if (HAS_HI) g1 = wmma4(ai_hi, w.bi01, g1);
  // xn' = -2log2e*(x[n]+b_ih[n]) ; hn' = -2log2e*(h[n]+b_hh[n])
  v8f xn = wmma4(ai_lo, w.bi10, splat8(w.cbi1));
  if (HAS_HI) xn = wmma4(ai_hi, w.bi11, xn);
  v8f hn = wmma4(ah_lo, w.bh10, wmma4(ah_hi, w.bh11, splat8(w.cbh1)));

  v8f hnew;
#pragma unroll
  for (int c = 0; c < 8; ++c) {
    float s  = sigm_s(g1[c]);                // r on cols 0-7, z on cols 8-15
    float zz = lane_xor8(s);                 // bring z to the r columns
    float n  = tanh_s(__builtin_fmaf(s, hn[c], xn[c]));
    hnew[c] = __builtin_fmaf(zz, h[c] - n, n);   // (1-z)*n + z*h
  }
  return hnew;
}

// D-layout -> LDS, one 16-float row per hidden unit j (0..7), indexed by batch row m (0..15)
// Store side: 2 x ds_store_b128 per contributing lane.
static __device__ inline void storeH(float* __restrict__ lds, v8f h, int lane) {
  if ((lane & 15) < 8) {
    int j  = lane & 15;
    int mb = (lane >> 4) * 8;
    float* p = lds + j * 16 + mb;
    float4 lo4 = {h[0], h[1], h[2], h[3]};
    float4 hi4 = {h[4], h[5], h[6], h[7]};
    *(float4*)p       = lo4;
    *(float4*)(p + 4) = hi4;
  }
}
// LDS (row per unit j, column per batch row m) -> A operand pair (K-groups 0..3 and 4..7); pairs fuse to
// ds_load_2addr_b32 (dword offsets {0,16} and {64,80} off a common base).
static __device__ inline void loadA(const float* __restrict__ lds, int lane,
                                    v2f* lo, v2f* hi) {
  int m  = lane & 15;
  int kb = (lane >> 4) * 2;
  const float* p = lds + kb * 16 + m;
  v2f l; l.x = p[0];   l.y = p[16];     // K = kb, kb+1
  v2f h; h.x = p[64];  h.y = p[80];     // K = kb+4, kb+5
  *lo = l; *hi = h;
}

__global__ __launch_bounds__(128) void gru3_fused_kernel(
    const float* __restrict__ x,
    const float* __restrict__ Wih0, const float* __restrict__ Whh0,
    const float* __restrict__ bih0, const float* __restrict__ bhh0,
    const float* __restrict__ Wih1, const float* __restrict__ Whh1,
    const float* __restrict__ bih1, const float* __restrict__ bhh1,
    const float* __restrict__ Wih2, const float* __restrict__ Whh2,
    const float* __restrict__ bih2, const float* __restrict__ bhh2,
    const float* __restrict__ fcW, const float* __restrict__ fcb,
    float* __restrict__ out) {
  __shared__ float smem[4 * 3 * 128];          // 4 waves * 3 layers * 8x16 floats
  const int lane = threadIdx.x & 31;
  const int wid  = threadIdx.x >> 5;
  const int tile = blockIdx.x * 4 + wid;       // 16-batch tile per wave
  const int b0   = tile * 16;

  float* lds0 = &smem[wid * 384];
  float* lds1 = lds0 + 128;
  float* lds2 = lds0 + 256;
  for (int i = lane; i < 384; i += 32) lds0[i] = 0.f;   // h(t=0) = 0
  wait_ds();

  const LayerW w0 = make_layer(Wih0, FIN, Whh0, bih0, bhh0, lane);
  const LayerW w1 = make_layer(Wih1, HID, Whh1, bih1, bhh1, lane);
  const LayerW w2 = make_layer(Wih2, HID, Whh2, bih2, bhh2, lane);

  const v8f vz = {0.f,0.f,0.f,0.f,0.f,0.f,0.f,0.f};
  v8f h0 = vz, h1 = vz, h2 = vz, acc = vz;

  // software-pipelined x load: ax holds x(t); prefetch x(t+1) at loop top.
  // All lanes load from a valid address (lane&15); upper half zeroed by cndmask
  // (avoids a divergent exec-mask branch every iteration).
  const float* xp = x + (size_t)b0 * FIN;
  v2f ax;
  {
    v2f raw = *(const v2f*)(xp + (lane & 15) * FIN);
    ax.x = (lane < 16) ? raw.x : 0.f;
    ax.y = (lane < 16) ? raw.y : 0.f;
  }

  for (int t = 0; t < TSTEPS; ++t) {
    const float* xnp = (t < TSTEPS - 1) ? (xp + (size_t)BATCH * FIN) : xp;
    v2f ax_next;
    {
      v2f raw = *(const v2f*)(xnp + (lane & 15) * FIN);
      ax_next.x = (lane < 16) ? raw.x : 0.f;
      ax_next.y = (lane < 16) ? raw.y : 0.f;
    }

    // ---- layer 0 (input K=2, fits K-group 0) ----
    v2f a0l, a0h; loadA(lds0, lane, &a0l, &a0h);
    v2f az = {0.f, 0.f};
    h0 = gru_step<false>(w0, ax, az, a0l, a0h, h0);
    storeH(lds0, h0, lane);
    wait_ds();

    // ---- layer 1 ----
    v2f i1l, i1h; loadA(lds0, lane, &i1l, &i1h);
    v2f a1l, a1h; loadA(lds1, lane, &a1l, &a1h);
    h1 = gru_step<true>(w1, i1l, i1h, a1l, a1h, h1);
    storeH(lds1, h1, lane);
    wait_ds();

    // ---- layer 2 ----
    v2f i2l, i2h; loadA(lds1, lane, &i2l, &i2h);
    v2f a2l, a2h; loadA(lds2, lane, &a2l, &a2h);
    h2 = gru_step<true>(w2, i2l, i2h, a2l, a2h, h2);
    storeH(lds2, h2, lane);
    acc += h2;
    wait_ds();

    ax = ax_next;
    xp = xnp;
  }

  // ---- mean over T, then FC (OUT=4) ----
  v8f mean = acc * (1.f / (float)TSTEPS);
  storeH(lds0, mean, lane);
  wait_ds();
  if (lane < 16) {
    float4 o;
    float* po = &o.x;
#pragma unroll
    for (int oo = 0; oo < NOUT; ++oo) {
      float a2 = fcb[oo];
#pragma unroll
      for (int j = 0; j < HID; ++j) a2 += lds0[j * 16 + lane] * fcW[oo * HID + j];
      po[oo] = a2;
    }
    *(float4*)(out + (size_t)(b0 + lane) * NOUT) = o;
  }
}

extern "C" void kernel_launch(void* const* d_in, const int* in_sizes, int n_in,
                              void* d_out, int out_size, void* d_ws, size_t ws_size,
                              hipStream_t stream) {
  (void)in_sizes; (void)n_in; (void)out_size; (void)d_ws; (void)ws_size;
  const float* x    = (const float*)d_in[0];
  const float* Wih0 = (const float*)d_in[1];
  const float* Whh0 = (const float*)d_in[2];
  const float* bih0 = (const float*)d_in[3];
  const float* bhh0 = (const float*)d_in[4];
  const float* Wih1 = (const float*)d_in[5];
  const float* Whh1 = (const float*)d_in[6];
  const float* bih1 = (const float*)d_in[7];
  const float* bhh1 = (const float*)d_in[8];
  const float* Wih2 = (const float*)d_in[9];
  const float* Whh2 = (const float*)d_in[10];
  const float* bih2 = (const float*)d_in[11];
  const float* bhh2 = (const float*)d_in[12];
  const float* fcW  = (const float*)d_in[13];
  const float* fcb  = (const float*)d_in[14];
  float* out = (float*)d_out;

  dim3 grid(BATCH / 64);   // 64 batch rows per block (4 waves * 16)
  dim3 block(128);
  gru3_fused_kernel<<<grid, block, 0, stream>>>(
      x, Wih0, Whh0, bih0, bhh0, Wih1, Whh1, bih1, bhh1,
      Wih2, Whh2, bih2, bhh2, fcW, fcb, out);
}